// SoftMultiFramePredictor_76441827935000
// MI455X (gfx1250) — compile-verified
//
#include <hip/hip_runtime.h>
#include <math.h>

typedef __attribute__((ext_vector_type(2))) float v2f;
typedef __attribute__((ext_vector_type(8))) float v8f;

#define NF 2
#define CH 3
#define HH 96
#define WW 160
#define HW (HH * WW)
#define KS 31
#define TT 15
#define KK 961          // 31*31 taps per pixel
#define NCHUNK 240      // K=4 chunks covering taps 0..959; tap 960 in epilogue

// d_out layout (floats, concatenated in reference return order):
//   out:         [NF][CH][HH][WW]   offset 0        (92160)
//   positions:   [NF][HH][WW][2]    offset 92160    (61440)
//   exp_norm:    [NF][HH][WW]       offset 153600   (30720)
//   norm_factor: [NF][HH][WW]       offset 184320   (30720)

__global__ __launch_bounds__(256) void smfp_kernel(
    const float* __restrict__ frames,   // [NF][CH][HH][WW]
    const float* __restrict__ weights,  // [NF][HH][WW][KS][KS]
    float* __restrict__ out)
{
    // B-fragment table: [chunk][lane][2] floats = 240*32*2*4 = 60 KB.
    // Identical for every wave; built once per block. Reused post-loop as the
    // WMMA D-fragment spill area (after a barrier).
    __shared__ float btab[NCHUNK * 64];

    const int tid = threadIdx.x;

    // ---- build the constant B table (cols: 0=gy, 1=gx, 2=|g|, 3=1, 4..15=0) ----
    for (int idx = tid; idx < NCHUNK * 32; idx += 256) {
        const int ln = idx & 31;
        const int pp = ln & 15;
        const int t0 = (idx >> 5) * 4 + 2 * (ln >> 4);   // tap for K-row 2*half
        int k = t0 / KS, l = t0 - k * KS;
        float gx = (float)(k - TT), gy = (float)(l - TT);
        float v0 = (pp == 0) ? gy : (pp == 1) ? gx
                 : (pp == 2) ? sqrtf(gx * gx + gy * gy)
                 : (pp == 3) ? 1.0f : 0.0f;
        const int t1 = t0 + 1;                           // max 959, never wraps past end
        k = t1 / KS; l = t1 - k * KS;
        gx = (float)(k - TT); gy = (float)(l - TT);
        float v1 = (pp == 0) ? gy : (pp == 1) ? gx
                 : (pp == 2) ? sqrtf(gx * gx + gy * gy)
                 : (pp == 3) ? 1.0f : 0.0f;
        btab[idx * 2 + 0] = v0;
        btab[idx * 2 + 1] = v1;
    }
    __syncthreads();

    const int wave = tid >> 5;
    const int lane = tid & 31;
    const int p    = lane & 15;   // A row (pixel) AND B column index
    const int half = lane >> 4;   // supplies K pair {0,1} vs {2,3} per chunk

    // 16-pixel groups along rows: 2*96*10 = 1920 groups = 240 blocks * 8 waves
    const int g  = blockIdx.x * 8 + wave;
    const int n  = g / 960;
    const int r  = g % 960;
    const int i  = r / 10;
    const int j  = (r % 10) * 16 + p;

    const float* wbase = weights + ((size_t)((n * HH + i) * WW + j)) * KK;
    const float* xb    = frames + n * CH * HW;   // wave-uniform -> SGPR base
    const int jm = j - TT;                        // loop-invariant

    // per-lane tap walker: tap t = 4*c + 2*half
    int t   = 2 * half;
    int l0  = t;            // t % 31 (t starts < 31)
    int ipu = i - TT;       // unclamped row i + k0 - TT
    int ju  = jm + l0;      // unclamped col j + l0 - TT

    v8f acc = {};
    float o0 = 0.f, o1 = 0.f, o2 = 0.f;

    for (int c = 0; c < NCHUNK; ++c) {
        // ---- stream two contiguous weights, exponentiate once ----
        float w0 = wbase[t];
        float w1 = wbase[t + 1];
        __builtin_prefetch(wbase + t + 128, 0, 1);   // global_prefetch_b8
        float e0 = __expf(w0);
        float e1 = __expf(w1);

        // ---- WMMA accumulate against the precomputed B fragment ----
        v2f bv = *(const v2f*)(btab + c * 64 + lane * 2);  // ds_load_b64
        v2f a; a.x = e0; a.y = e1;
        acc = __builtin_amdgcn_wmma_f32_16x16x4_f32(
                  false, a, false, bv, (short)0, acc, false, false);

        // ---- per-pixel patch dot products (frames are cache-resident) ----
        int wrap1 = (l0 == KS - 1);                 // tap t+1 starts a new row
        int ip0 = min(max(ipu, 0), HH - 1);
        int ip1 = wrap1 ? min(max(ipu + 1, 0), HH - 1) : ip0;
        int jp0 = min(max(ju, 0), WW - 1);
        int jp1 = min(max(wrap1 ? jm : (ju + 1), 0), WW - 1);
        int off0 = ip0 * WW + jp0;
        int off1 = ip1 * WW + jp1;
        o0 = fmaf(e0, xb[off0],          fmaf(e1, xb[off1],          o0));
        o1 = fmaf(e0, xb[off0 + HW],     fmaf(e1, xb[off1 + HW],     o1));
        o2 = fmaf(e0, xb[off0 + 2*HW],   fmaf(e1, xb[off1 + 2*HW],   o2));

        // ---- advance 4 taps (branchless) ----
        t += 4;
        l0 += 4;
        int wr = (l0 >= KS);
        l0  = wr ? l0 - KS : l0;
        ipu += wr;
        ju  += wr ? (4 - KS) : 4;
    }

    // ---- epilogue: tap 960 (k=30,l=30); only K-row 0 (half==0) is live ----
    {
        float wlast = wbase[KK - 1];                // valid for all lanes
        float e0 = (half == 0) ? __expf(wlast) : 0.0f;
        float b0 = (p == 0) ? 15.f : (p == 1) ? 15.f
                 : (p == 2) ? 21.2132034f : (p == 3) ? 1.0f : 0.0f;
        v2f a; a.x = e0; a.y = 0.f;
        v2f b; b.x = b0; b.y = 0.f;
        acc = __builtin_amdgcn_wmma_f32_16x16x4_f32(
                  false, a, false, b, (short)0, acc, false, false);
        if (half == 0) {
            int off = min(i + TT, HH - 1) * WW + min(j + TT, WW - 1);
            o0 = fmaf(e0, xb[off],          o0);
            o1 = fmaf(e0, xb[off + HW],     o1);
            o2 = fmaf(e0, xb[off + 2*HW],   o2);
        }
    }

    // ---- cross-half reduce the channel sums (wave32) ----
    o0 += __shfl_xor(o0, 16);
    o1 += __shfl_xor(o1, 16);
    o2 += __shfl_xor(o2, 16);

    // ---- spill D fragment (reuse btab region; all waves done reading it) ----
    __syncthreads();
    float* sp = btab + wave * 256;     // 32 lanes * 8 floats per wave
#pragma unroll
    for (int rr = 0; rr < 8; ++rr)
        sp[lane * 8 + rr] = acc[rr];
    // same-wave LDS store->load is in-order; no barrier needed for own region

    if (half == 0) {
        // D layout: value(M,N) lives in VGPR (M&7) of lane N + 16*(M>>3)
        const int sl = 16 * (p >> 3);
        const int vr = p & 7;
        float sgy = sp[(sl + 0) * 8 + vr];   // sum e*(l-15) -> positions[...,0]
        float sgx = sp[(sl + 1) * 8 + vr];   // sum e*(k-15) -> positions[...,1]
        float sen = sp[(sl + 2) * 8 + vr];   // sum e*|g|    -> exp_norm
        float snf = sp[(sl + 3) * 8 + vr];   // sum e        -> norm_factor

        const size_t pix = ((size_t)n * HH + i) * WW + j;
        float* out_o  = out;                                // [NF][CH][HH][WW]
        float* out_p  = out + (size_t)NF * CH * HW;         // [NF][HH][WW][2]
        float* out_en = out_p + (size_t)NF * HW * 2;        // [NF][HH][WW]
        float* out_nf = out_en + (size_t)NF * HW;           // [NF][HH][WW]

        out_o[((size_t)(n * CH + 0) * HH + i) * WW + j] = o0;
        out_o[((size_t)(n * CH + 1) * HH + i) * WW + j] = o1;
        out_o[((size_t)(n * CH + 2) * HH + i) * WW + j] = o2;
        out_p[pix * 2 + 0] = sgy;
        out_p[pix * 2 + 1] = sgx;
        out_en[pix] = sen;
        out_nf[pix] = snf;
    }
}

extern "C" void kernel_launch(void* const* d_in, const int* in_sizes, int n_in,
                              void* d_out, int out_size, void* d_ws, size_t ws_size,
                              hipStream_t stream) {
    const float* frames  = (const float*)d_in[0];   // (1, 2, 3, 96, 160) fp32
    const float* weights = (const float*)d_in[1];   // (1, 2, 96, 160, 31, 31) fp32
    float* out = (float*)d_out;                     // 215040 fp32, tuple concat

    smfp_kernel<<<dim3(240), dim3(256), 0, stream>>>(frames, weights, out);
}